// Sampler_23210003268199
// MI455X (gfx1250) — compile-verified
//
#include <hip/hip_runtime.h>

typedef float v2f __attribute__((ext_vector_type(2)));
typedef float v8f __attribute__((ext_vector_type(8)));

#define D_DIM   128
#define KSEL    16
#define MAXDEG  96
#define NEGV    (-1.0e30f)

// ---------------------------------------------------------------------------
// Exact JAX threefry2x32 uniform for key(42): u_i in [1e-9, 1)
// ---------------------------------------------------------------------------
__device__ __forceinline__ unsigned int tf_rotl(unsigned int x, int d) {
  return (x << d) | (x >> (32 - d));
}

__device__ __forceinline__ float jax_uniform_u(unsigned int i, unsigned int n) {
  const unsigned int half = (n + 1u) >> 1;          // n even here; general form
  const unsigned int idx  = (i < half) ? i : (i - half);
  unsigned int x0 = idx;
  unsigned int x1 = idx + half;
  const unsigned int ks0 = 0u;                       // key(42) -> (0, 42)
  const unsigned int ks1 = 42u;
  const unsigned int ks2 = ks0 ^ ks1 ^ 0x1BD11BDAu;
  x0 += ks0; x1 += ks1;
#define TF4(R0, R1, R2, R3)                                   \
  x0 += x1; x1 = tf_rotl(x1, R0); x1 ^= x0;                   \
  x0 += x1; x1 = tf_rotl(x1, R1); x1 ^= x0;                   \
  x0 += x1; x1 = tf_rotl(x1, R2); x1 ^= x0;                   \
  x0 += x1; x1 = tf_rotl(x1, R3); x1 ^= x0;
  TF4(13, 15, 26, 6)  x0 += ks1; x1 += ks2 + 1u;
  TF4(17, 29, 16, 24) x0 += ks2; x1 += ks0 + 2u;
  TF4(13, 15, 26, 6)  x0 += ks0; x1 += ks1 + 3u;
  TF4(17, 29, 16, 24) x0 += ks1; x1 += ks2 + 4u;
  TF4(13, 15, 26, 6)  x0 += ks2; x1 += ks0 + 5u;
#undef TF4
  const unsigned int bits = (i < half) ? x0 : x1;
  float f = __uint_as_float((bits >> 9) | 0x3f800000u) - 1.0f;
  f = f + 1e-9f;                    // (maxval-minval)==1.0f in fp32, + minval
  return fmaxf(1e-9f, f);
}

// ---------------------------------------------------------------------------
// Kernel 1: CSR row pointers (starts[N] = E)
// ---------------------------------------------------------------------------
__global__ __launch_bounds__(256) void starts_kernel(
    const int* __restrict__ row, int* __restrict__ starts, int Nn, int Ee) {
  const int r = blockIdx.x * 256 + threadIdx.x;
  if (r > Nn) return;
  if (r == Nn) { starts[Nn] = Ee; return; }
  int lo = 0, hi = Ee;
  while (lo < hi) {
    const int mid = (lo + hi) >> 1;
    if (row[mid] < r) lo = mid + 1; else hi = mid;
  }
  starts[r] = lo;
}

// ---------------------------------------------------------------------------
// Kernel 2: per-node squared norms, one wave32 per node (float4 + shfl reduce)
// ---------------------------------------------------------------------------
__global__ __launch_bounds__(256) void norm_kernel(
    const float* __restrict__ x, float* __restrict__ norm2, int Nn) {
  const int lane = threadIdx.x & 31;
  const int wid  = threadIdx.x >> 5;
  const int node = blockIdx.x * 8 + wid;
  if (node >= Nn) return;
  const float4 v = *(const float4*)(x + (size_t)node * D_DIM + lane * 4);
  float s = v.x * v.x + v.y * v.y + v.z * v.z + v.w * v.w;
#pragma unroll
  for (int off = 16; off >= 1; off >>= 1) s += __shfl_xor(s, off, 32);
  if (lane == 0) norm2[node] = s;
}

// ---------------------------------------------------------------------------
// Kernel 3: Gumbel scores. One wave32 per 16 edges; A = x[row] tile (16x128),
// B = x[col] tile (128x16); Gram via 32x V_WMMA_F32_16X16X4_F32; keep diagonal.
// ---------------------------------------------------------------------------
__global__ __launch_bounds__(256) void score_kernel(
    const float* __restrict__ x, const int* __restrict__ row,
    const int* __restrict__ col, const float* __restrict__ norm2,
    float* __restrict__ scores, unsigned int Ee) {
  const int lane = threadIdx.x & 31;
  const int wid  = threadIdx.x >> 5;
  const unsigned int eb = ((unsigned int)blockIdx.x * 8u + (unsigned int)wid) * 16u;
  if (eb >= Ee) return;                    // wave-uniform; EXEC stays full

  const int m  = lane & 15;                // edge-in-tile this lane serves
  const int kk = lane >> 4;                // K half select per ISA A/B layout
  unsigned int em = eb + (unsigned int)m;
  if (em >= Ee) em = Ee - 1u;              // clamp tail (loads stay valid)
  const int rm = row[em];
  const int cm = col[em];

  // A(16x4) lane layout: VGPR0=K(2*kk), VGPR1=K(2*kk+1) for row m.
  // B(4x16) lane layout mirrors it with col n = lane&15.
  const float* pa = x + (size_t)rm * D_DIM + 2 * kk;
  const float* pb = x + (size_t)cm * D_DIM + 2 * kk;

  v8f acc = {0.f, 0.f, 0.f, 0.f, 0.f, 0.f, 0.f, 0.f};
#pragma unroll
  for (int c = 0; c < D_DIM / 4; ++c) {
    const v2f a = *(const v2f*)(pa + 4 * c);
    const v2f b = *(const v2f*)(pb + 4 * c);
    acc = __builtin_amdgcn_wmma_f32_16x16x4_f32(
        /*neg_a=*/false, a, /*neg_b=*/false, b,
        /*c_mod=*/(short)0, acc, /*reuse_a=*/false, /*reuse_b=*/false);
  }

  // Extract diagonal: edge e<8 at (vgpr e, lane e); e>=8 at (vgpr e-8, lane e+16).
  float dot = 0.0f;
#pragma unroll
  for (int v = 0; v < 8; ++v) {
    const float dlo = __shfl(acc[v], v, 32);
    const float dhi = __shfl(acc[v], v + 24, 32);
    if (lane == v)     dot = dlo;
    if (lane == v + 8) dot = dhi;
  }

  const unsigned int e = eb + (unsigned int)lane;
  if (lane < 16 && e < Ee) {
    const float ss = fmaxf(norm2[rm] + norm2[cm] - 2.0f * dot, 0.0f);
    const float w  = expf(-ss) + 1e-6f;
    const float u  = jax_uniform_u(e, Ee);
    scores[e] = logf(w) - logf(-logf(u));
  }
}

// ---------------------------------------------------------------------------
// Kernel 4: per-node stable top-16 (lax.top_k tie-break: lower index first).
// Outputs concatenated [src | dst | mask] as floats.
// ---------------------------------------------------------------------------
__global__ __launch_bounds__(256) void topk_kernel(
    const int* __restrict__ col, const float* __restrict__ scores,
    const int* __restrict__ starts, float* __restrict__ out, int Nn) {
  const int r = blockIdx.x * 256 + threadIdx.x;
  if (r >= Nn) return;
  const int s0 = starts[r];
  int deg = starts[r + 1] - s0;
  if (deg > MAXDEG) deg = MAXDEG;          // overflow edges hit the dump column

  float bv[KSEL];
  int   bc[KSEL];
#pragma unroll
  for (int k = 0; k < KSEL; ++k) { bv[k] = NEGV; bc[k] = -1; }

  for (int t = 0; t < deg; ++t) {
    const float sc = scores[s0 + t];
    if (sc > bv[KSEL - 1]) {               // strict: boundary tie keeps earlier idx
      float cv = sc;
      int   ci = col[s0 + t];
#pragma unroll
      for (int k = 0; k < KSEL; ++k) {     // insert at first strictly-greater slot
        if (cv > bv[k]) {
          const float tv = bv[k]; const int ti = bc[k];
          bv[k] = cv; bc[k] = ci;
          cv = tv;    ci = ti;
        }
      }
    }
  }

  const size_t nk   = (size_t)Nn * KSEL;
  const size_t base = (size_t)r * KSEL;
  float* __restrict__ osrc = out;
  float* __restrict__ odst = out + nk;
  float* __restrict__ omsk = out + 2 * nk;
#pragma unroll
  for (int k = 0; k < KSEL; ++k) {
    osrc[base + k] = (float)r;
    odst[base + k] = (float)bc[k];
    omsk[base + k] = (bv[k] > NEGV * 0.5f) ? 1.0f : 0.0f;
  }
}

// ---------------------------------------------------------------------------
extern "C" void kernel_launch(void* const* d_in, const int* in_sizes, int n_in,
                              void* d_out, int out_size, void* d_ws, size_t ws_size,
                              hipStream_t stream) {
  const float* x   = (const float*)d_in[0];
  const int*   row = (const int*)d_in[1];
  const int*   col = (const int*)d_in[2];
  // d_in[3] = num_samples (16); KSEL is the compile-time selection width.

  const int Nn = in_sizes[0] / D_DIM;
  const int Ee = in_sizes[1];
  float* out = (float*)d_out;

  // Workspace: [scores: E f32][starts: N+1 i32][norm2: N f32]  (~13.3 MB)
  float* scores = (float*)d_ws;
  int*   starts = (int*)((char*)d_ws + (size_t)Ee * sizeof(float));
  float* norm2  = (float*)((char*)d_ws + (size_t)Ee * sizeof(float)
                                       + (size_t)(Nn + 1) * sizeof(int));

  starts_kernel<<<(Nn + 256) / 256, 256, 0, stream>>>(row, starts, Nn, Ee);
  norm_kernel<<<(Nn + 7) / 8, 256, 0, stream>>>(x, norm2, Nn);
  score_kernel<<<(Ee + 127) / 128, 256, 0, stream>>>(x, row, col, norm2, scores,
                                                     (unsigned int)Ee);
  topk_kernel<<<(Nn + 255) / 256, 256, 0, stream>>>(col, scores, starts, out, Nn);
}